// GCN_128849019395
// MI455X (gfx1250) — compile-verified
//
#include <hip/hip_runtime.h>
#include <hip/hip_bf16.h>

typedef float v2f __attribute__((ext_vector_type(2)));
typedef float v8f __attribute__((ext_vector_type(8)));

#define HID 64
#define NGRAPH 256
#define NCLASS 10

// ---------------- degree / norm ----------------
__global__ void deg_init_kernel(float* deg, int n) {
    int i = blockIdx.x * blockDim.x + threadIdx.x;
    if (i < n) deg[i] = 1.0f;                      // self-loop contributes 1
}

__global__ void deg_count_kernel(const int* __restrict__ dst, float* deg, int ne) {
    int e = blockIdx.x * blockDim.x + threadIdx.x;
    if (e < ne) atomicAdd(&deg[dst[e]], 1.0f);
}

__global__ void deg_inv_kernel(const float* __restrict__ deg, float* dinv, int n) {
    int i = blockIdx.x * blockDim.x + threadIdx.x;
    if (i < n) dinv[i] = rsqrtf(deg[i]);           // deg >= 1 always
}

// ---------------- WMMA GEMM: out[N,64] = (relu?)(A[N,64]) @ W[64,64] -------
// One wave computes a 16-row x 64-col tile via V_WMMA_F32_16X16X4_F32.
// W staged in LDS (16 KB). 8 waves / block -> 128 rows per block.
// RELU is a compile-time flag so the hot loop has no per-element selects.
template <int RELU>
__global__ void __launch_bounds__(256)
gemm64_wmma_kernel(const float* __restrict__ A, const float* __restrict__ W,
                   float* __restrict__ out, int nrows) {
    __shared__ float Wl[HID * HID];
    const int tid = threadIdx.x;
    for (int i = tid; i < HID * HID; i += 256) Wl[i] = W[i];
    __syncthreads();

    const int wave = tid >> 5;
    const int lane = tid & 31;
    const int half = lane >> 4;      // 0: K pair low, 1: K pair high
    const int m    = lane & 15;

    int tile = blockIdx.x * 8 + wave;
    const int ntiles = nrows >> 4;   // nrows is a multiple of 16
    if (tile >= ntiles) tile = ntiles - 1;   // duplicate last tile (benign)
    const int row0 = tile << 4;

    v8f acc0 = {}, acc1 = {}, acc2 = {}, acc3 = {};

    const float* arow = A + (size_t)(row0 + m) * HID;

    #pragma unroll
    for (int kk = 0; kk < 16; ++kk) {
        const int kbase = kk * 4 + 2 * half;       // even -> 8B aligned
        v2f a = *(const v2f*)(arow + kbase);
        if (RELU) { a.x = fmaxf(a.x, 0.0f); a.y = fmaxf(a.y, 0.0f); }

        // B frags: W[kbase..kbase+1][nt*16 + m] from LDS
        v2f b0, b1, b2, b3;
        const float* wr0 = &Wl[kbase * HID];
        const float* wr1 = &Wl[(kbase + 1) * HID];
        b0.x = wr0[m];      b0.y = wr1[m];
        b1.x = wr0[16 + m]; b1.y = wr1[16 + m];
        b2.x = wr0[32 + m]; b2.y = wr1[32 + m];
        b3.x = wr0[48 + m]; b3.y = wr1[48 + m];

        acc0 = __builtin_amdgcn_wmma_f32_16x16x4_f32(false, a, false, b0, (short)0, acc0, false, false);
        acc1 = __builtin_amdgcn_wmma_f32_16x16x4_f32(false, a, false, b1, (short)0, acc1, false, false);
        acc2 = __builtin_amdgcn_wmma_f32_16x16x4_f32(false, a, false, b2, (short)0, acc2, false, false);
        acc3 = __builtin_amdgcn_wmma_f32_16x16x4_f32(false, a, false, b3, (short)0, acc3, false, false);
    }

    // C/D layout: VGPR j <-> row (j + 8*half), col = lane&15 within each 16-col tile
    #pragma unroll
    for (int j = 0; j < 8; ++j) {
        float* orow = out + (size_t)(row0 + j + 8 * half) * HID + m;
        orow[0]  = acc0[j];
        orow[16] = acc1[j];
        orow[32] = acc2[j];
        orow[48] = acc3[j];
    }
}

// ---------------- self-loop + bias init: out = tmp * dinv[i]^2 + b[f] ------
__global__ void selfloop_bias_kernel(const float* __restrict__ tmp,
                                     const float* __restrict__ dinv,
                                     const float* __restrict__ b,
                                     float* __restrict__ out, int n) {
    int t = blockIdx.x * blockDim.x + threadIdx.x;
    if (t >= n * HID) return;
    int i = t >> 6, f = t & 63;
    float dv = dinv[i];
    out[t] = tmp[t] * dv * dv + b[f];
}

// ---------------- edge scatter: out[dst] += h[src] * dinv[src]*dinv[dst] ---
// wave-per-edge: lane f covers features 2f,2f+1 -> one coalesced b64 load per
// lane (wave reads whole 256B row in one instruction), then 2 f32 atomics.
__global__ void edge_scatter_kernel(const int* __restrict__ src,
                                    const int* __restrict__ dst,
                                    const float* __restrict__ dinv,
                                    const float* __restrict__ h,
                                    float* __restrict__ out, int ne) {
    int t = blockIdx.x * blockDim.x + threadIdx.x;
    int e = t >> 5, lane = t & 31;
    if (e >= ne) return;
    int s = src[e], d = dst[e];
    float nrm = dinv[s] * dinv[d];
    const v2f hv = *(const v2f*)(h + (size_t)s * HID + 2 * lane);
    float* od = out + (size_t)d * HID + 2 * lane;
    atomicAdd(&od[0], hv.x * nrm);
    atomicAdd(&od[1], hv.y * nrm);
}

// ---------------- pooling ----------------
__global__ void pool_zero_kernel(float* sums, float* counts) {
    int t = blockIdx.x * blockDim.x + threadIdx.x;
    if (t < NGRAPH * HID) sums[t] = 0.0f;
    if (t < NGRAPH) counts[t] = 0.0f;
}

__global__ void pool_accum_kernel(const float* __restrict__ h,
                                  const int* __restrict__ batch,
                                  float* sums, float* counts, int n) {
    int t = blockIdx.x * blockDim.x + threadIdx.x;
    int i = t >> 5, lane = t & 31;
    if (i >= n) return;
    int g = batch[i];
    const v2f hv = *(const v2f*)(h + (size_t)i * HID + 2 * lane);
    float* sg = sums + (size_t)g * HID + 2 * lane;
    atomicAdd(&sg[0], hv.x);
    atomicAdd(&sg[1], hv.y);
    if (lane == 0) atomicAdd(&counts[g], 1.0f);
}

__global__ void head_kernel(const float* __restrict__ sums,
                            const float* __restrict__ counts,
                            const float* __restrict__ Wlin,
                            const float* __restrict__ blin,
                            float* __restrict__ out) {
    int t = blockIdx.x * blockDim.x + threadIdx.x;
    if (t >= NGRAPH * NCLASS) return;
    int g = t / NCLASS, c = t % NCLASS;
    float inv = 1.0f / fmaxf(counts[g], 1.0f);
    float accv = blin[c];
    #pragma unroll
    for (int f = 0; f < HID; ++f)
        accv += (sums[g * HID + f] * inv) * Wlin[f * NCLASS + c];
    out[t] = accv;
}

// ---------------- host-side launcher ----------------
extern "C" void kernel_launch(void* const* d_in, const int* in_sizes, int n_in,
                              void* d_out, int out_size, void* d_ws, size_t ws_size,
                              hipStream_t stream) {
    const float* x     = (const float*)d_in[0];
    const int*   eidx  = (const int*)d_in[1];
    const int*   batch = (const int*)d_in[2];
    const float* W1 = (const float*)d_in[3];  const float* b1 = (const float*)d_in[4];
    const float* W2 = (const float*)d_in[5];  const float* b2 = (const float*)d_in[6];
    const float* W3 = (const float*)d_in[7];  const float* b3 = (const float*)d_in[8];
    const float* Wlin = (const float*)d_in[9]; const float* blin = (const float*)d_in[10];
    float* out = (float*)d_out;

    const int N = in_sizes[0] / HID;        // 100000
    const int E = in_sizes[1] / 2;          // 1250000
    const int* src = eidx;
    const int* dst = eidx + E;

    // workspace carve-up (floats); offsets keep 8B alignment for v2f loads
    float* ws    = (float*)d_ws;
    float* deg   = ws;                       // N
    float* dinv  = ws + N;                   // N
    float* bufA  = dinv + N;                 // N*HID
    float* bufB  = bufA + (size_t)N * HID;   // N*HID
    float* sums  = bufB + (size_t)N * HID;   // 256*64
    float* cnts  = sums + NGRAPH * HID;      // 256

    const int T = 256;
    const int gN   = (N + T - 1) / T;
    const int gE   = (E + T - 1) / T;
    const int gNH  = (N * HID + T - 1) / T;
    const int gE32 = ((E * 32) + T - 1) / T;
    const int gN32 = ((N * 32) + T - 1) / T;
    const int ntiles = N / 16;
    const int gGemm  = (ntiles + 7) / 8;     // 8 waves (16-row tiles) per block

    // normalization
    deg_init_kernel<<<gN, T, 0, stream>>>(deg, N);
    deg_count_kernel<<<gE, T, 0, stream>>>(dst, deg, E);
    deg_inv_kernel<<<gN, T, 0, stream>>>(deg, dinv, N);

    // layer 1: x @ W1 -> A ; B = A*selfnorm + b1 ; scatter A into B
    gemm64_wmma_kernel<0><<<gGemm, T, 0, stream>>>(x, W1, bufA, N);
    selfloop_bias_kernel<<<gNH, T, 0, stream>>>(bufA, dinv, b1, bufB, N);
    edge_scatter_kernel<<<gE32, T, 0, stream>>>(src, dst, dinv, bufA, bufB, E);

    // layer 2: relu(B) @ W2 -> A ; B = A*selfnorm + b2 ; scatter A into B
    gemm64_wmma_kernel<1><<<gGemm, T, 0, stream>>>(bufB, W2, bufA, N);
    selfloop_bias_kernel<<<gNH, T, 0, stream>>>(bufA, dinv, b2, bufB, N);
    edge_scatter_kernel<<<gE32, T, 0, stream>>>(src, dst, dinv, bufA, bufB, E);

    // layer 3: relu(B) @ W3 -> A ; B = A*selfnorm + b3 ; scatter A into B
    gemm64_wmma_kernel<1><<<gGemm, T, 0, stream>>>(bufB, W3, bufA, N);
    selfloop_bias_kernel<<<gNH, T, 0, stream>>>(bufA, dinv, b3, bufB, N);
    edge_scatter_kernel<<<gE32, T, 0, stream>>>(src, dst, dinv, bufA, bufB, E);

    // mean-pool + head
    pool_zero_kernel<<<(NGRAPH * HID + T - 1) / T, T, 0, stream>>>(sums, cnts);
    pool_accum_kernel<<<gN32, T, 0, stream>>>(bufB, batch, sums, cnts, N);
    head_kernel<<<(NGRAPH * NCLASS + T - 1) / T, T, 0, stream>>>(sums, cnts, Wlin, blin, out);
}